// HNM_heatmap_32521492365973
// MI455X (gfx1250) — compile-verified
//
#include <hip/hip_runtime.h>
#include <stdint.h>
#include <stddef.h>

// ---------------- problem constants ----------------
#define ROWS   64            // B*C
#define NPR    262144        // 64^3 elements per row
#define NB     8192          // t-value buckets per row
#define TILE   1024          // TDM staging tile (floats)
#define BPR    64            // blocks per row for streaming passes
#define SLICE  (NPR / BPR)   // 4096 elements per block
#define THREADS 256
#define DEFSEL 1000u

// ---------------- small helpers ----------------
__device__ __forceinline__ uint32_t fkey(float f) {
  // monotone (ascending) 32-bit key for floats
  uint32_t b = __float_as_uint(f);
  return (b & 0x80000000u) ? ~b : (b | 0x80000000u);
}
__device__ __forceinline__ float sl1(float p, float t) {
  float d = p - t, ad = fabsf(d);
  return ad < 1.f ? 0.5f * d * d : ad - 0.5f;
}
// Distribution-matched, exactly monotone (non-decreasing in t) bucket for
// negative t in (-1.001, -0.001): -t uniform -> near-uniform bucket load ~30.
__device__ __forceinline__ uint32_t tbucket(float tv) {
  float v = (-tv) * 8175.0f;          // in (8.2, 8183.2); RTE mul is monotone
  uint32_t q = (uint32_t)v;           // floor (v >= 0)
  if (q > NB - 1u) q = NB - 1u;
  return (NB - 1u) - q;               // ascending t -> ascending bucket
}

// ---------------- CDNA5 TDM path (6-arg clang-23 builtin on this toolchain) --
#if defined(__has_builtin)
#if __has_builtin(__builtin_amdgcn_tensor_load_to_lds) && \
    __has_builtin(__builtin_amdgcn_s_wait_tensorcnt)
#define HAVE_TDM 1
#endif
#endif

#if defined(HAVE_TDM)
typedef unsigned int tdm_v4u __attribute__((ext_vector_type(4)));
typedef int          tdm_v8i __attribute__((ext_vector_type(8)));
typedef int          tdm_v4i __attribute__((ext_vector_type(4)));

// DMA nElems fp32 from global (gptr) into LDS byte offset lds_off.
// 1-row 2D tile: tile_dim0 = tensor_dim0 = nElems, data_size = 4B.
// D# group0 (128b) + group1 (256b) describe the tile; groups 2/3 and the
// trailing int32x8 operand are zero (tile_dim2/3/4 unused).
__device__ __forceinline__ void tdm_load_tile(uint32_t lds_off,
                                              const void* gptr,
                                              uint32_t nElems) {
  uint64_t ga = (uint64_t)(uintptr_t)gptr;
  tdm_v4u g0;
  g0[0] = 1u;                                            // count=1, user D#
  g0[1] = lds_off;                                       // lds_addr
  g0[2] = (uint32_t)ga;                                  // global_addr[31:0]
  g0[3] = (uint32_t)((ga >> 32) & 0x01FFFFFFull) | (2u << 30); // addr[56:32] | type=2
  tdm_v8i g1;
  g1[0] = (int)(2u << 16);           // wg_mask=0 (not in cluster), data_size=4B
  g1[1] = (int)(nElems << 16);       // tensor_dim0[15:0] @bits63:48
  g1[2] = (int)((nElems >> 16) | (1u << 16)); // tensor_dim0[31:16], tensor_dim1=1
  g1[3] = (int)(nElems << 16);       // tile_dim0 @bits127:112
  g1[4] = 1;                         // tile_dim1=1 (tile_dim2=0 unused)
  g1[5] = (int)nElems;               // tensor_dim0_stride[31:0]
  g1[6] = (int)(nElems << 16);       // tensor_dim1_stride[15:0] (benign)
  g1[7] = 0;
  tdm_v4i gz4; gz4[0] = 0; gz4[1] = 0; gz4[2] = 0; gz4[3] = 0;
  tdm_v8i gz8;
  gz8[0] = 0; gz8[1] = 0; gz8[2] = 0; gz8[3] = 0;
  gz8[4] = 0; gz8[5] = 0; gz8[6] = 0; gz8[7] = 0;
  __builtin_amdgcn_tensor_load_to_lds(g0, g1, gz4, gz4, gz8, 0);
}
#endif

// =====================================================================
// Pass 1: per-row sel count, positive-loss partials, and t-bucket histogram.
// TDM double-buffers 4KB tiles of h and t into LDS while waves do LDS
// histogram atomics on the previous tile.
// =====================================================================
__global__ void k_pass1(const float* __restrict__ h, const float* __restrict__ t,
                        uint32_t* __restrict__ hist_all,
                        uint32_t* __restrict__ selArr,
                        float* __restrict__ posPart,
                        int r0) {
  const int slot = blockIdx.y;
  const int row  = r0 + slot;
  const float* __restrict__ hrow = h + (size_t)row * NPR;
  const float* __restrict__ trow = t + (size_t)row * NPR;
  uint32_t* __restrict__ hist = hist_all + (size_t)slot * NB;

  __shared__ uint32_t lhist[NB];      // 32 KB
  __shared__ float    redf[THREADS];  // 1 KB
  __shared__ uint32_t redu[THREADS];  // 1 KB
#if defined(HAVE_TDM)
  __shared__ float bT[2][TILE];       // 8 KB
  __shared__ float bH[2][TILE];       // 8 KB
#endif

  const int tid = threadIdx.x;
  for (int i = tid; i < NB; i += THREADS) lhist[i] = 0;
  __syncthreads();

  const int start = blockIdx.x * SLICE;
  uint32_t mySel = 0;
  float    myPos = 0.f;

#if defined(HAVE_TDM)
  const int wid = __builtin_amdgcn_readfirstlane((int)(threadIdx.x >> 5));
  const int ntiles = SLICE / TILE;
  if (wid == 0) {  // scalar-uniform branch: only wave 0 drives the TDM
    tdm_load_tile((uint32_t)(uintptr_t)&bT[0][0], trow + start, TILE);
    tdm_load_tile((uint32_t)(uintptr_t)&bH[0][0], hrow + start, TILE);
  }
  for (int tt = 0; tt < ntiles; ++tt) {
    const int cur = tt & 1;
    if (wid == 0) {
      if (tt + 1 < ntiles) {
        tdm_load_tile((uint32_t)(uintptr_t)&bT[cur ^ 1][0],
                      trow + start + (tt + 1) * TILE, TILE);
        tdm_load_tile((uint32_t)(uintptr_t)&bH[cur ^ 1][0],
                      hrow + start + (tt + 1) * TILE, TILE);
        __builtin_amdgcn_s_wait_tensorcnt(2);  // tile tt's two DMAs done
      } else {
        __builtin_amdgcn_s_wait_tensorcnt(0);
      }
    }
    __syncthreads();
    volatile const float* pT = &bT[cur][0];  // TDM writes are invisible to AA
    volatile const float* pH = &bH[cur][0];
    for (int e = tid; e < TILE; e += THREADS) {
      float tv = pT[e], hv = pH[e];
      if (tv < 0.f) {
        atomicAdd(&lhist[tbucket(tv)], 1u);
      } else {
        mySel++; myPos += sl1(hv, tv);
      }
    }
    __syncthreads();
  }
#else
  for (int i = start + tid; i < start + SLICE; i += THREADS) {
    float tv = trow[i], hv = hrow[i];
    if (tv < 0.f) atomicAdd(&lhist[tbucket(tv)], 1u);
    else { mySel++; myPos += sl1(hv, tv); }
  }
  __syncthreads();
#endif

  for (int i = tid; i < NB; i += THREADS) {
    uint32_t v = lhist[i];
    if (v) atomicAdd(&hist[i], v);
  }
  // deterministic block reductions
  redf[tid] = myPos; redu[tid] = mySel;
  __syncthreads();
  for (int s = THREADS / 2; s > 0; s >>= 1) {
    if (tid < s) { redf[tid] += redf[tid + s]; redu[tid] += redu[tid + s]; }
    __syncthreads();
  }
  if (tid == 0) {
    posPart[slot * BPR + blockIdx.x] = redf[0];
    atomicAdd(&selArr[row], redu[0]);   // integer atomic: order-independent
  }
}

// =====================================================================
// Exclusive prefix over the 8192 buckets of one row; base + scatter cursor.
// =====================================================================
__global__ void k_prefix(const uint32_t* __restrict__ hist_all,
                         uint32_t* __restrict__ base_all,
                         uint32_t* __restrict__ cursor_all) {
  const int slot = blockIdx.x;
  const uint32_t* hist = hist_all + (size_t)slot * NB;
  uint32_t* base = base_all + (size_t)slot * NB;
  uint32_t* cur  = cursor_all + (size_t)slot * NB;
  __shared__ uint32_t part[THREADS];
  const int tid = threadIdx.x;
  const int per = NB / THREADS;  // 32
  uint32_t s = 0;
  for (int k = 0; k < per; ++k) s += hist[tid * per + k];
  part[tid] = s;
  __syncthreads();
  if (tid == 0) {
    uint32_t run = 0;
    for (int i = 0; i < THREADS; ++i) { uint32_t v = part[i]; part[i] = run; run += v; }
  }
  __syncthreads();
  uint32_t run = part[tid];
  for (int k = 0; k < per; ++k) {
    int idx = tid * per + k;
    base[idx] = run; cur[idx] = run;
    run += hist[idx];
  }
}

// =====================================================================
// Counting-sort scatter of full 32-bit t-keys by bucket (negatives only).
// =====================================================================
__global__ void k_scatter(const float* __restrict__ t,
                          uint32_t* __restrict__ cursor_all,
                          uint32_t* __restrict__ skeys_all,
                          int r0) {
  const int slot = blockIdx.y;
  const int row  = r0 + slot;
  const float* __restrict__ trow = t + (size_t)row * NPR;
  uint32_t* __restrict__ cur = cursor_all + (size_t)slot * NB;
  uint32_t* __restrict__ sk  = skeys_all + (size_t)slot * NPR;
  const int start = blockIdx.x * SLICE;
  for (int i = start + threadIdx.x; i < start + SLICE; i += THREADS) {
    __builtin_prefetch(trow + i + 2048, 0, 1);   // global_prefetch_b8
    float tv = trow[i];
    if (tv < 0.f) {
      uint32_t u = fkey(tv);
      uint32_t p = atomicAdd(&cur[tbucket(tv)], 1u);
      sk[p] = u;
    }
  }
}

// =====================================================================
// Radix-select init / per-round pass / per-round finalize.
// =====================================================================
__global__ void k_rowinit(const uint32_t* __restrict__ selArr,
                          uint32_t* __restrict__ KvArr,
                          uint32_t* __restrict__ prefixArr,
                          uint32_t* __restrict__ kremArr, int r0, int Gc) {
  int i = threadIdx.x;
  if (i < Gc) {
    int row = r0 + i;
    uint32_t s = selArr[row];
    uint32_t K = s > 0 ? s : DEFSEL;
    KvArr[row] = K; kremArr[row] = K; prefixArr[row] = 0u;
  }
}

__global__ void k_rsel(const float* __restrict__ h, const float* __restrict__ t,
                       uint32_t* __restrict__ rshist,
                       const uint32_t* __restrict__ prefixArr,
                       int shift, int r0) {
  const int slot = blockIdx.y;
  const int row  = r0 + slot;
  const float* __restrict__ hrow = h + (size_t)row * NPR;
  const float* __restrict__ trow = t + (size_t)row * NPR;
  __shared__ uint32_t lh[256];
  const int tid = threadIdx.x;
  if (tid < 256) lh[tid] = 0;
  __syncthreads();
  const uint32_t pref = prefixArr[row];
  const int start = blockIdx.x * SLICE;
  for (int i = start + tid; i < start + SLICE; i += THREADS) {
    __builtin_prefetch(hrow + i + 2048, 0, 1);
    float tv = trow[i];
    if (tv < 0.f) {
      uint32_t key = fkey(hrow[i]);
      if ((((uint64_t)(key ^ pref)) >> (shift + 8)) == 0ull)   // prefix match
        atomicAdd(&lh[(key >> shift) & 255u], 1u);
    }
  }
  __syncthreads();
  if (tid < 256) {
    uint32_t v = lh[tid];
    if (v) atomicAdd(&rshist[(size_t)row * 256 + tid], v);
  }
}

__global__ void k_rsfin(uint32_t* __restrict__ rshist,
                        uint32_t* __restrict__ prefixArr,
                        uint32_t* __restrict__ kremArr,
                        int shift, int r0, int Gc) {
  int i = threadIdx.x;
  if (i >= Gc) return;
  int row = r0 + i;
  uint32_t* hh = rshist + (size_t)row * 256;
  uint32_t krem = kremArr[row];
  uint32_t cum = 0, chosen = 0;
  for (int d = 255; d >= 0; --d) {
    uint32_t c = hh[d];
    if (cum + c >= krem) { chosen = (uint32_t)d; break; }
    cum += c;
  }
  kremArr[row]   = krem - cum;
  prefixArr[row] |= chosen << shift;
  for (int d = 0; d < 256; ++d) hh[d] = 0;   // ready for next round/chunk
}

// =====================================================================
// Final pass: for each selected negative (h-key >= threshold), compute its
// ascending-t rank j among negatives, gather h[j],t[j], accumulate smooth-L1.
// =====================================================================
__global__ void k_neg(const float* __restrict__ h, const float* __restrict__ t,
                      const uint32_t* __restrict__ hist_all,
                      const uint32_t* __restrict__ base_all,
                      const uint32_t* __restrict__ skeys_all,
                      const uint32_t* __restrict__ prefixArr,
                      float* __restrict__ negPart, int r0) {
  const int slot = blockIdx.y;
  const int row  = r0 + slot;
  const float* __restrict__ hrow = h + (size_t)row * NPR;
  const float* __restrict__ trow = t + (size_t)row * NPR;
  const uint32_t* __restrict__ hist = hist_all + (size_t)slot * NB;
  const uint32_t* __restrict__ base = base_all + (size_t)slot * NB;
  const uint32_t* __restrict__ sk   = skeys_all + (size_t)slot * NPR;
  const uint32_t thresh = prefixArr[row];
  __shared__ float red[THREADS];
  const int tid = threadIdx.x;
  float acc = 0.f;
  const int start = blockIdx.x * SLICE;
  for (int i = start + tid; i < start + SLICE; i += THREADS) {
    __builtin_prefetch(trow + i + 2048, 0, 1);
    float tv = trow[i];
    if (tv < 0.f) {
      uint32_t hk = fkey(hrow[i]);
      if (hk >= thresh) {                       // exactly K selected (tie-free)
        uint32_t u  = fkey(tv);
        uint32_t b  = tbucket(tv);
        uint32_t s0 = base[b], c = hist[b];
        uint32_t less = 0;
        for (uint32_t k = 0; k < c; ++k) less += (sk[s0 + k] < u) ? 1u : 0u;
        uint32_t j = s0 + less;                 // rank in ascending-t order
        acc += sl1(hrow[j], trow[j]);           // bug-faithful flat index use
      }
    }
  }
  red[tid] = acc;
  __syncthreads();
  for (int s = THREADS / 2; s > 0; s >>= 1) {
    if (tid < s) red[tid] += red[tid + s];
    __syncthreads();
  }
  if (tid == 0) negPart[slot * BPR + blockIdx.x] = red[0];
}

__global__ void k_sumparts(const float* __restrict__ posPart,
                           const float* __restrict__ negPart,
                           float* __restrict__ possumArr,
                           float* __restrict__ negsumArr, int r0, int Gc) {
  int i = threadIdx.x;
  if (i >= Gc) return;
  int row = r0 + i;
  float ps = 0.f, ns = 0.f;
  for (int b = 0; b < BPR; ++b) { ps += posPart[i * BPR + b]; ns += negPart[i * BPR + b]; }
  possumArr[row] = ps; negsumArr[row] = ns;
}

__global__ void k_final(const uint32_t* __restrict__ selArr,
                        const float* __restrict__ possumArr,
                        const uint32_t* __restrict__ KvArr,
                        const float* __restrict__ negsumArr,
                        float* __restrict__ out) {
  if (threadIdx.x == 0 && blockIdx.x == 0) {
    float total = 0.f;
    for (int r = 0; r < ROWS; ++r) {
      uint32_t s = selArr[r];
      float pos = (s > 0) ? possumArr[r] / (float)s : 0.f;
      total += pos + negsumArr[r] / (float)KvArr[r];
    }
    out[0] = total / (float)ROWS;
  }
}

// =====================================================================
// Host launcher: workspace layout + row-chunked pipeline (all on `stream`).
// =====================================================================
extern "C" void kernel_launch(void* const* d_in, const int* in_sizes, int n_in,
                              void* d_out, int out_size, void* d_ws, size_t ws_size,
                              hipStream_t stream) {
  const float* h = (const float*)d_in[0];
  const float* t = (const float*)d_in[1];
  float* out = (float*)d_out;

  uint8_t* w = (uint8_t*)d_ws;
  size_t off = 0;
  auto take = [&](size_t bytes) {
    size_t o = off;
    off = (off + bytes + 255) & ~(size_t)255;
    return o;
  };
  uint32_t* selArr    = (uint32_t*)(w + take(ROWS * 4));
  uint32_t* KvArr     = (uint32_t*)(w + take(ROWS * 4));
  uint32_t* prefixArr = (uint32_t*)(w + take(ROWS * 4));
  uint32_t* kremArr   = (uint32_t*)(w + take(ROWS * 4));
  float*    possumArr = (float*)   (w + take(ROWS * 4));
  float*    negsumArr = (float*)   (w + take(ROWS * 4));
  uint32_t* rshist    = (uint32_t*)(w + take((size_t)ROWS * 256 * 4));
  float*    posPart   = (float*)   (w + take((size_t)ROWS * BPR * 4));
  float*    negPart   = (float*)   (w + take((size_t)ROWS * BPR * 4));
  size_t smallEnd = off;
  (void)hipMemsetAsync(w, 0, smallEnd, stream);

  // big per-row-slot arrays: hist, base, cursor (NB u32 each) + skeys (NPR u32)
  size_t perSlot = (size_t)NB * 4 * 3 + (size_t)NPR * 4;   // ~1.15 MB/row
  size_t avail = (ws_size > smallEnd) ? (ws_size - smallEnd) : 0;
  int G = (int)(avail / perSlot);
  if (G < 1) G = 1;
  if (G > ROWS) G = ROWS;
  uint32_t* hist_all   = (uint32_t*)(w + smallEnd);
  uint32_t* base_all   = hist_all   + (size_t)G * NB;
  uint32_t* cursor_all = base_all   + (size_t)G * NB;
  uint32_t* skeys_all  = cursor_all + (size_t)G * NB;

  for (int r0 = 0; r0 < ROWS; r0 += G) {
    int Gc = ((ROWS - r0) < G) ? (ROWS - r0) : G;
    (void)hipMemsetAsync(hist_all, 0, (size_t)Gc * NB * 4, stream);
    dim3 grid(BPR, Gc);
    k_pass1  <<<grid, THREADS, 0, stream>>>(h, t, hist_all, selArr, posPart, r0);
    k_prefix <<<Gc,   THREADS, 0, stream>>>(hist_all, base_all, cursor_all);
    k_scatter<<<grid, THREADS, 0, stream>>>(t, cursor_all, skeys_all, r0);
    k_rowinit<<<1,    ROWS,    0, stream>>>(selArr, KvArr, prefixArr, kremArr, r0, Gc);
    for (int rnd = 0; rnd < 4; ++rnd) {
      int shift = 24 - 8 * rnd;
      k_rsel <<<grid, THREADS, 0, stream>>>(h, t, rshist, prefixArr, shift, r0);
      k_rsfin<<<1,    ROWS,    0, stream>>>(rshist, prefixArr, kremArr, shift, r0, Gc);
    }
    k_neg     <<<grid, THREADS, 0, stream>>>(h, t, hist_all, base_all, skeys_all,
                                             prefixArr, negPart, r0);
    k_sumparts<<<1,    ROWS,    0, stream>>>(posPart, negPart, possumArr, negsumArr, r0, Gc);
  }
  k_final<<<1, 1, 0, stream>>>(selArr, possumArr, KvArr, negsumArr, out);

  (void)in_sizes; (void)n_in; (void)out_size;
}